// GraphConvLayer_17884243821392
// MI455X (gfx1250) — compile-verified
//
#include <hip/hip_runtime.h>

typedef __attribute__((ext_vector_type(2))) float v2f;
typedef __attribute__((ext_vector_type(8))) float v8f;

#define GH 64
#define GW 64
#define NNODE (GH * GW)       // 4096
#define CIN 128
#define COUT 128
#define BATCH 4
#define NROWS (BATCH * NNODE) // 16384
#define NEG_SLOPE 0.1f

// ---------------------------------------------------------------------------
// Adjacency: 4-NN on a 64x64 grid, symmetrized. All neighbors lie within a
// 5x5 Chebyshev window (4th-NN distance <= 2 even at corners), so each node's
// adjacency is a 25-bit window mask. Tie-break by linear index (stable).
// ---------------------------------------------------------------------------
__device__ __forceinline__ void top4(int r, int c, int* out_idx) {
    unsigned long long best[4] = {~0ull, ~0ull, ~0ull, ~0ull};
    for (int dr = -2; dr <= 2; ++dr) {
        for (int dc = -2; dc <= 2; ++dc) {
            if (dr == 0 && dc == 0) continue;
            int rr = r + dr, cc = c + dc;
            if (rr < 0 || rr >= GH || cc < 0 || cc >= GW) continue;
            unsigned d2  = (unsigned)(dr * dr + dc * dc);
            unsigned idx = (unsigned)(rr * GW + cc);
            unsigned long long key = ((unsigned long long)d2 << 32) | idx;
            for (int s = 0; s < 4; ++s) {
                if (key < best[s]) {
                    for (int t = 3; t > s; --t) best[t] = best[t - 1];
                    best[s] = key;
                    break;
                }
            }
        }
    }
    for (int s = 0; s < 4; ++s) out_idx[s] = (int)(best[s] & 0xffffffffu);
}

__global__ void build_masks(unsigned* __restrict__ masks) {
    int u = blockIdx.x * blockDim.x + threadIdx.x;
    if (u >= NNODE) return;
    int r = u / GW, c = u % GW;
    unsigned mask = 0;

    // forward edges: my 4 nearest neighbors
    int nn[4];
    top4(r, c, nn);
    for (int s = 0; s < 4; ++s) {
        int v = nn[s];
        int vr = v / GW, vc = v % GW;
        mask |= 1u << ((vr - r + 2) * 5 + (vc - c + 2));
    }
    // reverse edges: window candidates that picked me
    for (int dr = -2; dr <= 2; ++dr) {
        for (int dc = -2; dc <= 2; ++dc) {
            if (dr == 0 && dc == 0) continue;
            int vr = r + dr, vc = c + dc;
            if (vr < 0 || vr >= GH || vc < 0 || vc >= GW) continue;
            int vn[4];
            top4(vr, vc, vn);
            for (int s = 0; s < 4; ++s)
                if (vn[s] == u) mask |= 1u << ((dr + 2) * 5 + (dc + 2));
        }
    }
    masks[u] = mask;
}

// ---------------------------------------------------------------------------
// proj = x @ W via V_WMMA_F32_16X16X4_F32 (full fp32 precision).
// Block = 128 threads = 4 waves; each wave: 16 rows x 128 cols (8 C-tiles).
// W staged in LDS PRE-SWIZZLED into B-fragment order so that each lane's
// fragment is one aligned ds_load_b64 into an even VGPR pair (no movs):
//   element W[k][col] of a 32-row chunk -> Blds_flat[(k>>2)*512 + col*4 + (k&3)]
//   lane fragment (k4, nt, hi): v2f at Blds + (k4*128 + nt*16 + ln)*2 + hi
// Bank map: lanes 0-15 -> banks {0..61 step 4}(+1), lanes 16-31 -> {2..62}(+1)
// => all 64 banks, conflict-free.
//
// Fragment layouts (ISA 7.12.2, 32-bit):
//  A 16x4:  lanes 0-15 -> {K0,K1} of row M=lane; lanes 16-31 -> {K2,K3}
//  B 4x16:  lanes 0-15 -> {row K0, row K1} at N=lane; lanes 16-31 -> {K2,K3}
//  C 16x16: VGPR r -> M=r (lanes 0-15) / M=r+8 (lanes 16-31), N=lane&15
// ---------------------------------------------------------------------------
__global__ __launch_bounds__(128)
void gemm_wmma(const float* __restrict__ X, const float* __restrict__ Wm,
               float* __restrict__ P) {
    __shared__ v2f Blds[8 * COUT * 2];   // 16 KB: (k4, col, hi) -> float2

    const int lane = threadIdx.x & 31;
    const int wave = threadIdx.x >> 5;
    const int m0   = blockIdx.x * 64 + wave * 16;
    const int ln   = lane & 15;
    const int hi   = lane >> 4;          // 0: K0/K1 half, 1: K2/K3 half
    const float* __restrict__ arowp = X + (size_t)(m0 + ln) * CIN + hi * 2;

    v8f acc[8] = {};

    for (int kc = 0; kc < CIN; kc += 32) {
        __syncthreads();
        // stage + swizzle a 32x128 chunk of W into fragment order
        float* __restrict__ bf = (float*)Blds;
        for (int t = threadIdx.x; t < 32 * COUT; t += 128) {
            const int k = t >> 7, col = t & 127;               // coalesced read
            bf[((k >> 2) * COUT + col) * 4 + (k & 3)] =
                Wm[(size_t)(kc + k) * COUT + col];
        }
        __syncthreads();

        #pragma unroll
        for (int k4 = 0; k4 < 8; ++k4) {
            const v2f a = *(const v2f*)(arowp + kc + k4 * 4);  // 8B/lane
            const v2f* __restrict__ bp = Blds + (k4 * COUT + ln) * 2 + hi;
            #pragma unroll
            for (int nt = 0; nt < 8; ++nt) {
                const v2f b = bp[nt * 32];                     // ds_load_b64
                acc[nt] = __builtin_amdgcn_wmma_f32_16x16x4_f32(
                    /*neg_a=*/false, a, /*neg_b=*/false, b,
                    /*c_mod=*/(short)0, acc[nt],
                    /*reuse_a=*/false, /*reuse_b=*/false);
            }
        }
    }

    // store: row = m0 + rrow + hi*8, col = nt*16 + ln
    float* __restrict__ outp = P + (size_t)(m0 + hi * 8) * COUT + ln;
    #pragma unroll
    for (int nt = 0; nt < 8; ++nt) {
        #pragma unroll
        for (int rrow = 0; rrow < 8; ++rrow) {
            outp[(size_t)rrow * COUT + nt * 16] = acc[nt][rrow];
        }
    }
}

// ---------------------------------------------------------------------------
// out[b,u,:] = leaky_relu( sum_{v in adj(u)} proj[b,v,:] )
// One block per (batch, node); one thread per channel (coalesced 512B rows,
// proj stays L2-resident at 8MB).
// ---------------------------------------------------------------------------
__global__ __launch_bounds__(128)
void gather_lrelu(const float* __restrict__ P, const unsigned* __restrict__ masks,
                  float* __restrict__ out) {
    const int node = blockIdx.x & (NNODE - 1);
    const int b    = blockIdx.x >> 12;
    const int ch   = threadIdx.x;
    const int r = node >> 6, c = node & 63;

    unsigned m = masks[node];
    float s = 0.0f;
    while (m) {
        const int bit = __ffs(m) - 1;
        m &= (m - 1);
        const int dr = bit / 5 - 2;
        const int dc = bit % 5 - 2;
        const int v  = (r + dr) * GW + (c + dc);
        s += P[((size_t)b * NNODE + v) * COUT + ch];
    }
    out[((size_t)b * NNODE + node) * COUT + ch] = (s >= 0.0f) ? s : NEG_SLOPE * s;
}

// ---------------------------------------------------------------------------
extern "C" void kernel_launch(void* const* d_in, const int* in_sizes, int n_in,
                              void* d_out, int out_size, void* d_ws, size_t ws_size,
                              hipStream_t stream) {
    const float* x  = (const float*)d_in[0];   // (4,64,64,128) f32
    const float* w  = (const float*)d_in[1];   // (128,128) f32
    float* out      = (float*)d_out;           // (4,64,64,128) f32

    unsigned* masks = (unsigned*)d_ws;                                  // 16 KB
    float*    proj  = (float*)((char*)d_ws + NNODE * sizeof(unsigned)); // 8 MB

    build_masks<<<(NNODE + 127) / 128, 128, 0, stream>>>(masks);
    gemm_wmma<<<NROWS / 64, 128, 0, stream>>>(x, w, proj);
    gather_lrelu<<<BATCH * NNODE, 128, 0, stream>>>(proj, masks, out);
}